// AnchorHead_21371757265385
// MI455X (gfx1250) — compile-verified
//
#include <hip/hip_runtime.h>
#include <cstdint>
#include <math.h>

// ---------------- problem constants ----------------
#define NLVL   5
#define NB     8
#define NA     9
#define NC     80
#define NCH    (NA*NC)      // 720 channels per cls tensor
#define KSEL   500
#define NTOT   196416       // sum over levels of H*H*9 (per batch)
#define MAXRATIO 4.135166556742356f
#define IMGS   1024.0f

// per-level H*W and per-batch key offsets
// H = {128,64,32,16,8}; HW = {16384,4096,1024,256,64}; N = HW*9
// LOFF = {0,147456,184320,193536,195840}

__device__ __forceinline__ int lvl_of(int off) {
  return (off < 147456) ? 0 : (off < 184320) ? 1 : (off < 193536) ? 2 : (off < 195840) ? 3 : 4;
}
__device__ __forceinline__ int loff_of(int l) {
  return (l == 0) ? 0 : (l == 1) ? 147456 : (l == 2) ? 184320 : (l == 3) ? 193536 : 195840;
}
__device__ __forceinline__ unsigned mono_key(float f) {
  unsigned u = __float_as_uint(f);
  return (u & 0x80000000u) ? ~u : (u | 0x80000000u);
}

// ---------------- CDNA5 TDM availability ----------------
#if defined(__HIP_DEVICE_COMPILE__) && defined(__has_builtin)
#if __has_builtin(__builtin_amdgcn_tensor_load_to_lds) && __has_builtin(__builtin_amdgcn_s_wait_tensorcnt)
#define USE_TDM 1
#endif
#endif
#ifndef USE_TDM
#define USE_TDM 0
#endif

typedef unsigned int u32x4 __attribute__((ext_vector_type(4)));
typedef int          i32x8 __attribute__((ext_vector_type(8)));
typedef int          i32x4 __attribute__((ext_vector_type(4)));

#define TDM_CH 20   // channels per TDM tile (80/20 = 4 tiles per anchor group)

#if USE_TDM
// Issue one TDM load: tile = TDM_CH rows x tile0 elements of f32, row stride = HW elems.
// D# layout per CDNA5 ISA.
// clang-23 builtin arity is 6: groups 0..3 (4+8+4+4 dwords) + unused VADDR4 slot + cpol.
__device__ __forceinline__ void tdm_issue(unsigned long long gaddr, unsigned ldsaddr,
                                          int tile0, int HW) {
  u32x4 g0; i32x8 g1; i32x4 g2; i32x4 g3; i32x8 g4;
  g2[0]=0; g2[1]=0; g2[2]=0; g2[3]=0;
  g3[0]=0; g3[1]=0; g3[2]=0; g3[3]=0;
  g4[0]=0; g4[1]=0; g4[2]=0; g4[3]=0; g4[4]=0; g4[5]=0; g4[6]=0; g4[7]=0;
  // group0: [1:0]=count=1, [63:32]=lds_addr, [120:64]=global_addr, [127:126]=type=2
  g0[0] = 1u;
  g0[1] = ldsaddr;
  g0[2] = (unsigned)(gaddr & 0xFFFFFFFFull);
  g0[3] = (unsigned)((gaddr >> 32) & 0x01FFFFFFull) | (2u << 30);
  // group1
  unsigned td0 = (unsigned)tile0;        // tensor_dim0 (elements)
  unsigned td1 = (unsigned)TDM_CH;       // tensor_dim1 (lines)
  unsigned tile_d0 = (unsigned)tile0;
  unsigned tile_d1 = (unsigned)TDM_CH;
  g1[0] = 0x20000;                                        // data_size=2 (4B); wg_mask=0
  g1[1] = (int)((td0 & 0xFFFFu) << 16);                   // tensor_dim0[15:0] @ bits 63:48
  g1[2] = (int)(((td0 >> 16) & 0xFFFFu) | ((td1 & 0xFFFFu) << 16));
  g1[3] = (int)(((td1 >> 16) & 0xFFFFu) | (tile_d0 << 16));       // tile_dim0
  g1[4] = (int)(tile_d1 & 0xFFFFu);                        // tile_dim1 (tile_dim2=0 -> 2D)
  g1[5] = (int)((unsigned)HW);                             // tensor_dim0_stride lo32 (elems)
  g1[6] = 0;                                               // stride hi16 | dim1_stride lo16
  g1[7] = 0;
  __builtin_amdgcn_tensor_load_to_lds(g0, g1, g2, g3, g4, 0);
}
#endif

// ---------------- kernel 1: per-(batch,pos,anchor) max over 80 classes ----------------
// grid: ( ceil(HW/256), NB ), block 256
__global__ __launch_bounds__(256)
void maxcls_kernel(const float* __restrict__ cls, unsigned* __restrict__ mk,
                   int HW, int loff, int tile0) {
#if USE_TDM
  __shared__ float buf[2][TDM_CH * 256];
  const int b = blockIdx.y, t = threadIdx.x;
  const int pos0 = blockIdx.x * 256;
  const unsigned lds0 = (unsigned)(uintptr_t)&buf[0][0];
  const unsigned lds1 = (unsigned)(uintptr_t)&buf[1][0];
  const unsigned long long gbase =
      (unsigned long long)(uintptr_t)cls +
      ((unsigned long long)b * NCH * (unsigned long long)HW +
       (unsigned long long)pos0) * 4ull;
  const unsigned long long tstep = (unsigned long long)TDM_CH * (unsigned long long)HW * 4ull;
  // TDM ignores EXEC, so an exec-masked (branchless) guard would make all 8
  // waves issue duplicate DMAs. readfirstlane yields an SGPR -> scalar compare
  // -> a real s_cbranch, so only wave 0 reaches the tensor ops at all.
  const int wave_first_tid = __builtin_amdgcn_readfirstlane((int)threadIdx.x);
  const bool issuer = (wave_first_tid == 0);
  if (issuer) tdm_issue(gbase, lds0, tile0, HW);          // prologue: tile 0
  int tt = 0;
  for (int a = 0; a < NA; ++a) {
    float m = -3.4e38f;
    for (int sub = 0; sub < 4; ++sub, ++tt) {
      if (issuer) {
        if (tt + 1 < 36) {
          tdm_issue(gbase + (unsigned long long)(tt + 1) * tstep,
                    ((tt + 1) & 1) ? lds1 : lds0, tile0, HW);
          __builtin_amdgcn_s_wait_tensorcnt(1);           // tile tt complete
        } else {
          __builtin_amdgcn_s_wait_tensorcnt(0);
        }
      }
      __syncthreads();                                    // tile tt visible to all waves
      if (t < tile0) {
        const float* bb = &buf[tt & 1][0];
        #pragma unroll
        for (int r = 0; r < TDM_CH; ++r) m = fmaxf(m, bb[r * tile0 + t]);
      }
      __syncthreads();                                    // done reading before reuse
    }
    if (t < tile0) {
      int pos = pos0 + t;
      mk[(size_t)b * NTOT + loff + (size_t)pos * NA + a] = mono_key(m);
    }
  }
#else
  (void)tile0;
  const int b = blockIdx.y;
  const int pos = blockIdx.x * 256 + threadIdx.x;
  if (pos >= HW) return;
  const float* base = cls + (size_t)b * NCH * HW + pos;
  for (int a = 0; a < NA; ++a) {
    float m = -3.4e38f;
    const float* p = base + (size_t)a * NC * HW;
    #pragma unroll 8
    for (int c = 0; c < NC; ++c) m = fmaxf(m, p[(size_t)c * HW]);
    mk[(size_t)b * NTOT + loff + (size_t)pos * NA + a] = mono_key(m);
  }
#endif
}

// ---------------- kernel 2: init selection state ----------------
__global__ void init_kernel(unsigned* hist, unsigned* prefix, int* remaining,
                            int* cnt_gt, int* cnt_eq) {
  int i = blockIdx.x * 256 + threadIdx.x;
  if (i < NLVL * NB * 256) hist[i] = 0;
  if (i < NLVL * NB) { prefix[i] = 0; remaining[i] = KSEL; cnt_gt[i] = 0; cnt_eq[i] = 0; }
}

// ---------------- kernel 3: radix-select histogram pass ----------------
// grid: (768, NB), block 256
__global__ void hist_pass(const unsigned* __restrict__ mk, unsigned* __restrict__ hist,
                          const unsigned* __restrict__ prefix, int shift, unsigned himask) {
  const int b = blockIdx.y;
  const int off = blockIdx.x * 256 + threadIdx.x;
  if (off >= NTOT) return;
  const int l = lvl_of(off);
  const int sb = l * NB + b;
  const unsigned key = mk[(size_t)b * NTOT + off];
  if (((key ^ prefix[sb]) & himask) == 0)
    atomicAdd(&hist[sb * 256 + ((key >> shift) & 0xFF)], 1u);
}

// ---------------- kernel 4: radix-select scan (one thread per selector) ----------------
__global__ void scan_pass(unsigned* hist, unsigned* prefix, int* remaining, int shift) {
  const int sb = threadIdx.x;
  if (sb >= NLVL * NB) return;
  int rem = remaining[sb];
  unsigned c = 0;
  unsigned* h = hist + sb * 256;
  int sel = 0;
  for (int d = 255; d >= 0; --d) {
    unsigned hv = h[d];
    if (c + hv >= (unsigned)rem) { sel = d; rem -= (int)c; break; }
    c += hv;
  }
  prefix[sb] |= ((unsigned)sel) << shift;
  remaining[sb] = rem;
  for (int d = 0; d < 256; ++d) h[d] = 0;   // clean for next pass
}

// ---------------- kernel 5: compact the top-K candidates ----------------
__global__ void compact_kernel(const unsigned* __restrict__ mk,
                               const unsigned* __restrict__ prefix,
                               const int* __restrict__ remaining,
                               int* cnt_gt, int* cnt_eq,
                               unsigned* __restrict__ ckey, int* __restrict__ cidx) {
  const int b = blockIdx.y;
  const int off = blockIdx.x * 256 + threadIdx.x;
  if (off >= NTOT) return;
  const int l = lvl_of(off);
  const int sb = l * NB + b;
  const unsigned key = mk[(size_t)b * NTOT + off];
  const unsigned kth = prefix[sb];
  if (key > kth) {
    int g = atomicAdd(&cnt_gt[sb], 1);
    ckey[sb * KSEL + g] = key;
    cidx[sb * KSEL + g] = off - loff_of(l);
  } else if (key == kth) {
    int rem = remaining[sb];
    int e = atomicAdd(&cnt_eq[sb], 1);
    if (e < rem) {
      int s = (KSEL - rem) + e;
      ckey[sb * KSEL + s] = key;
      cidx[sb * KSEL + s] = off - loff_of(l);
    }
  }
}

// ---------------- kernel 6: bitonic sort 500 (padded to 512) per selector ----------------
__global__ __launch_bounds__(256)
void sort_kernel(const unsigned* __restrict__ ckey, const int* __restrict__ cidx,
                 int* __restrict__ topk) {
  __shared__ unsigned sk[512];
  __shared__ int si[512];
  const int sb = blockIdx.x, t = threadIdx.x;
  for (int i = t; i < 512; i += 256) {
    if (i < KSEL) { sk[i] = ckey[sb * KSEL + i]; si[i] = cidx[sb * KSEL + i]; }
    else          { sk[i] = 0u;                  si[i] = 0x7FFFFFFF; }
  }
  __syncthreads();
  for (int k2 = 2; k2 <= 512; k2 <<= 1) {
    for (int j = k2 >> 1; j > 0; j >>= 1) {
      for (int i = t; i < 512; i += 256) {
        int ixj = i ^ j;
        if (ixj > i) {
          unsigned ka = sk[i], kb = sk[ixj];
          int ia = si[i], ib = si[ixj];
          bool b_first = (kb > ka) || (kb == ka && ib < ia);   // b precedes a
          bool a_first = (ka > kb) || (ka == kb && ia < ib);
          bool sw = ((i & k2) == 0) ? b_first : a_first;
          if (sw) { sk[i] = kb; sk[ixj] = ka; si[i] = ib; si[ixj] = ia; }
        }
      }
      __syncthreads();
    }
  }
  for (int i = t; i < KSEL; i += 256) topk[sb * KSEL + i] = si[i];
}

// ---------------- kernel 7: gather scores + decode boxes ----------------
// grid: (KSEL, NB), block 128 ; one launch per level
__global__ __launch_bounds__(128)
void decode_kernel(const float* __restrict__ cls, const float* __restrict__ bp,
                   const int* __restrict__ topk, float* __restrict__ out,
                   int l, int W, int HW, float stride) {
  const int j = blockIdx.x, b = blockIdx.y, t = threadIdx.x;
  const int sb = l * NB + b;
  const int n = topk[sb * KSEL + j];
  const int a = n % NA;
  const int pos = n / NA;
  const int row = b * (NLVL * KSEL) + l * KSEL + j;
  if (t < NC) {
    float v = cls[((size_t)b * NCH + (size_t)a * NC + t) * HW + pos];
    out[(size_t)(NB * NLVL * KSEL * 4) + (size_t)row * NC + t] = 1.f / (1.f + expf(-v));
  } else if (t == NC) {
    const int x = pos % W, y = pos / W;
    const float* d = bp + ((size_t)b * (NA * 4) + (size_t)a * 4) * HW + pos;
    const float dx = d[0];
    const float dy = d[(size_t)HW];
    float dw = d[2 * (size_t)HW];
    float dh = d[3 * (size_t)HW];
    dw = fminf(fmaxf(dw, -MAXRATIO), MAXRATIO);
    dh = fminf(fmaxf(dh, -MAXRATIO), MAXRATIO);
    const float wrv[3] = {1.4142135623730951f, 1.f, 0.7071067811865476f};
    const float hrv[3] = {0.7071067811865476f, 1.f, 1.4142135623730951f};
    const float scv[3] = {1.f, 1.2599210498948732f, 1.5874010519681994f};
    const int r = a / 3, s5 = a % 3;
    const float base = stride * 4.f;
    const float pw = base * wrv[r] * scv[s5];
    const float ph = base * hrv[r] * scv[s5];
    const float px = (float)x * stride, py = (float)y * stride;
    const float gx = px + pw * dx, gy = py + ph * dy;
    const float gw = pw * expf(dw), gh = ph * expf(dh);
    float* o = out + (size_t)row * 4;
    o[0] = fminf(fmaxf(gx - gw * 0.5f, 0.f), IMGS);
    o[1] = fminf(fmaxf(gy - gh * 0.5f, 0.f), IMGS);
    o[2] = fminf(fmaxf(gx + gw * 0.5f, 0.f), IMGS);
    o[3] = fminf(fmaxf(gy + gh * 0.5f, 0.f), IMGS);
  }
}

// ---------------- host launcher ----------------
extern "C" void kernel_launch(void* const* d_in, const int* in_sizes, int n_in,
                              void* d_out, int out_size, void* d_ws, size_t ws_size,
                              hipStream_t stream) {
  (void)in_sizes; (void)n_in; (void)out_size; (void)ws_size;
  static const int   HWs[NLVL]  = {16384, 4096, 1024, 256, 64};
  static const int   Ws[NLVL]   = {128, 64, 32, 16, 8};
  static const int   LOFF[NLVL] = {0, 147456, 184320, 193536, 195840};
  static const float STR[NLVL]  = {8.f, 16.f, 32.f, 64.f, 128.f};

  const float* cls[NLVL]; const float* bp[NLVL];
  for (int i = 0; i < NLVL; ++i) {
    cls[i] = (const float*)d_in[2 * i];
    bp[i]  = (const float*)d_in[2 * i + 1];
  }
  float* out = (float*)d_out;
  char* ws = (char*)d_ws;
  size_t cur = 0;
  auto take = [&](size_t bytes) -> char* {
    char* p = ws + cur;
    cur = (cur + bytes + 255) & ~(size_t)255;
    return p;
  };
  unsigned* mk     = (unsigned*)take((size_t)NB * NTOT * 4);
  unsigned* hist   = (unsigned*)take((size_t)NLVL * NB * 256 * 4);
  unsigned* prefix = (unsigned*)take((size_t)NLVL * NB * 4);
  int* remaining   = (int*)take((size_t)NLVL * NB * 4);
  int* cnt_gt      = (int*)take((size_t)NLVL * NB * 4);
  int* cnt_eq      = (int*)take((size_t)NLVL * NB * 4);
  unsigned* ckey   = (unsigned*)take((size_t)NLVL * NB * KSEL * 4);
  int* cidx        = (int*)take((size_t)NLVL * NB * KSEL * 4);
  int* topk        = (int*)take((size_t)NLVL * NB * KSEL * 4);

  init_kernel<<<(NLVL * NB * 256 + 255) / 256, 256, 0, stream>>>(
      hist, prefix, remaining, cnt_gt, cnt_eq);

  for (int l = 0; l < NLVL; ++l) {
    const int HW = HWs[l];
    const int tile0 = (HW < 256) ? HW : 256;
    dim3 grid((HW + 255) / 256, NB);
    maxcls_kernel<<<grid, 256, 0, stream>>>(cls[l], mk, HW, LOFF[l], tile0);
  }

  const dim3 gsel((NTOT + 255) / 256, NB);
  for (int p = 0; p < 4; ++p) {
    const int shift = 24 - 8 * p;
    const unsigned himask = (p == 0) ? 0u : (0xFFFFFFFFu << (shift + 8));
    hist_pass<<<gsel, 256, 0, stream>>>(mk, hist, prefix, shift, himask);
    scan_pass<<<1, 64, 0, stream>>>(hist, prefix, remaining, shift);
  }

  compact_kernel<<<gsel, 256, 0, stream>>>(mk, prefix, remaining, cnt_gt, cnt_eq,
                                           ckey, cidx);
  sort_kernel<<<NLVL * NB, 256, 0, stream>>>(ckey, cidx, topk);

  for (int l = 0; l < NLVL; ++l) {
    dim3 grid(KSEL, NB);
    decode_kernel<<<grid, 128, 0, stream>>>(cls[l], bp[l], topk, out,
                                            l, Ws[l], HWs[l], STR[l]);
  }
}